// ModelPart31_90305982366375
// MI455X (gfx1250) — compile-verified
//
#include <hip/hip_runtime.h>
#include <hip/hip_bf16.h>
#include <stdint.h>

#define N_BOXES   6144
#define MAX_OUT   300
#define IOU_TH    0.3f
#define SORT_M    8192           // next pow2 >= N
#define NBLK      192            // N/32 column blocks
#define NT16      384            // N/16 row tiles

typedef __attribute__((ext_vector_type(2))) float v2f;
typedef __attribute__((ext_vector_type(8))) float v8f;

__device__ __forceinline__ uint32_t ballot32(bool p) {
#if __has_builtin(__builtin_amdgcn_ballot_w32)
  return __builtin_amdgcn_ballot_w32(p);
#else
  return (uint32_t)__ballot(p);
#endif
}

// ---------------------------------------------------------------------------
// Kernel 1: stable descending argsort of scores (bitonic in LDS), gather boxes
// ---------------------------------------------------------------------------
__global__ __launch_bounds__(1024) void sort_kernel(const float* __restrict__ rois,
                                                    const float* __restrict__ scores,
                                                    float4* __restrict__ boxes_s,
                                                    float* __restrict__ area_s,
                                                    int* __restrict__ order) {
  __shared__ float key[SORT_M];
  __shared__ int   ind[SORT_M];
  const int tid = threadIdx.x;

  for (int i = tid; i < SORT_M; i += 1024) {
    key[i] = (i < N_BOXES) ? scores[i] : -__builtin_huge_valf();
    ind[i] = i;
  }
  __syncthreads();

  for (int k = 2; k <= SORT_M; k <<= 1) {
    for (int j = k >> 1; j > 0; j >>= 1) {
      for (int i = tid; i < SORT_M; i += 1024) {
        const int ixj = i ^ j;
        if (ixj > i) {
          float ka = key[i], kb = key[ixj];
          int   ia = ind[i], ib = ind[ixj];
          // strict total order: score descending, index ascending on ties
          bool aFirst   = (ka > kb) || (ka == kb && ia < ib);
          bool descHere = ((i & k) == 0);
          bool doSwap   = descHere ? (!aFirst) : aFirst;
          if (doSwap) { key[i] = kb; key[ixj] = ka; ind[i] = ib; ind[ixj] = ia; }
        }
      }
      __syncthreads();
    }
  }

  for (int i = tid; i < N_BOXES; i += 1024) {
    const int o = ind[i];
    order[i] = o;
    const float* r = rois + (size_t)o * 5;   // cols: [batch, c0, c1, c2, c3]
    const float a = r[1], b = r[2], c = r[3], d = r[4];
    boxes_s[i] = make_float4(a, b, c, d);
    area_s[i]  = (c - a) * (d - b);          // matches reference f32 ordering
  }
}

// ---------------------------------------------------------------------------
// Kernel 2: suppression bitmask.  One wave = 16 rows x 32 cols.
// Pairwise union term area[i]+area[j] computed via V_WMMA_F32_16X16X4_F32:
//   A row m = (area_row[m], 1, area_row[m], 1),  B col n = (1, area_col[n], 0, 0)
// -> D(m,n) = area_row[m] + area_col[n] exactly (single f32 rounding).
// ---------------------------------------------------------------------------
__global__ __launch_bounds__(32) void mask_kernel(const float4* __restrict__ boxes_s,
                                                  const float* __restrict__ area_s,
                                                  uint32_t* __restrict__ mask) {
  const int cb      = blockIdx.x;          // 32-wide column block
  const int rt      = blockIdx.y;          // 16-row tile
  const int rowBase = rt * 16;
  const int colBase = cb * 32;
  const int lane    = threadIdx.x;

  // Entire block is at or below the diagonal: no j > i bits possible.
  if (colBase + 31 <= rowBase) {
    if (lane < 16) mask[(size_t)(rowBase + lane) * NBLK + cb] = 0u;
    return;
  }

  const int l15 = lane & 15;
  const float4 rb      = boxes_s[rowBase + l15];   // row boxes in lanes 0..15 (dup hi)
  const float  rowArea = area_s[rowBase + l15];

  // A matrix: replicate (area,1) in both lane halves -> robust K-half mapping
  v2f A; A[0] = rowArea; A[1] = 1.0f;

  uint32_t rowMask[16];
#pragma unroll
  for (int t = 0; t < 16; ++t) rowMask[t] = 0u;

  const int hiOff = (lane >= 16) ? 8 : 0;

#pragma unroll
  for (int tile = 0; tile < 2; ++tile) {
    const int    colT    = colBase + tile * 16;
    const float4 cbx     = boxes_s[colT + l15];
    const float  colArea = area_s[colT + l15];

    v2f B;
    B[0] = (lane < 16) ? 1.0f    : 0.0f;
    B[1] = (lane < 16) ? colArea : 0.0f;

    v8f C = {};
#if __has_builtin(__builtin_amdgcn_wmma_f32_16x16x4_f32)
    C = __builtin_amdgcn_wmma_f32_16x16x4_f32(false, A, false, B,
                                              (short)0, C, false, false);
#endif

#pragma unroll
    for (int v = 0; v < 8; ++v) {
      const int t = v + hiOff;                 // row within tile for this element
      const float r0 = __shfl(rb.x, t, 32);
      const float r1 = __shfl(rb.y, t, 32);
      const float r2 = __shfl(rb.z, t, 32);
      const float r3 = __shfl(rb.w, t, 32);
      const float ih = fminf(r2, cbx.z) - fmaxf(r0, cbx.x);
      const float iw = fminf(r3, cbx.w) - fmaxf(r1, cbx.y);
      const float inter = fmaxf(ih, 0.0f) * fmaxf(iw, 0.0f);
#if __has_builtin(__builtin_amdgcn_wmma_f32_16x16x4_f32)
      const float s = C[v];                    // area_row + area_col from WMMA
#else
      const float s = __shfl(rowArea, t, 32) + colArea;
#endif
      const float uni = s - inter;
      const float iou = (uni > 0.0f) ? (inter / uni) : 0.0f;
      const uint32_t bal = ballot32(iou > IOU_TH);
      // bits 0..15 -> row v, bits 16..31 -> row v+8 (cols colT..colT+15)
      rowMask[v]     |= (bal & 0xFFFFu) << (tile * 16);
      rowMask[v + 8] |= (bal >> 16)     << (tile * 16);
    }
  }

  // Ballots are wave-uniform; lane 0 stores all 16 rows with j > i filtering.
  if (lane == 0) {
#pragma unroll
    for (int t = 0; t < 16; ++t) {
      const int i = rowBase + t;
      const int d = i - colBase;               // bit position of i in this block
      uint32_t valid;
      if (d < 0)        valid = 0xFFFFFFFFu;
      else if (d >= 31) valid = 0u;
      else              valid = 0xFFFFFFFFu << (d + 1);
      mask[(size_t)i * NBLK + cb] = rowMask[t] & valid;
    }
  }
}

// ---------------------------------------------------------------------------
// Kernel 3: sequential greedy sweep in a single wave32 (no barriers),
// removed-state = 192 words held as 6 registers per lane, broadcast via shfl.
// ---------------------------------------------------------------------------
__global__ __launch_bounds__(32) void reduce_kernel(const uint32_t* __restrict__ mask,
                                                    const int* __restrict__ order,
                                                    int* __restrict__ out) {
  const int lane = threadIdx.x;
  uint32_t r0 = 0, r1 = 0, r2 = 0, r3 = 0, r4 = 0, r5 = 0;
  __shared__ uint32_t rem[NBLK];

  for (int i = 0; i < N_BOXES; ++i) {
    const int w    = i >> 5;       // word index 0..191
    const int slot = w >> 5;       // uniform 0..5
    const int src  = w & 31;       // owner lane
    uint32_t word;
    switch (slot) {
      case 0:  word = (uint32_t)__shfl((int)r0, src, 32); break;
      case 1:  word = (uint32_t)__shfl((int)r1, src, 32); break;
      case 2:  word = (uint32_t)__shfl((int)r2, src, 32); break;
      case 3:  word = (uint32_t)__shfl((int)r3, src, 32); break;
      case 4:  word = (uint32_t)__shfl((int)r4, src, 32); break;
      default: word = (uint32_t)__shfl((int)r5, src, 32); break;
    }
    if (((word >> (i & 31)) & 1u) == 0u) {     // box i kept -> suppress its row
      const uint32_t* row = mask + (size_t)i * NBLK;
      r0 |= row[lane];
      r1 |= row[lane + 32];
      r2 |= row[lane + 64];
      r3 |= row[lane + 96];
      r4 |= row[lane + 128];
      r5 |= row[lane + 160];
    }
  }

  rem[lane]       = r0;  rem[lane + 32]  = r1;  rem[lane + 64]  = r2;
  rem[lane + 96]  = r3;  rem[lane + 128] = r4;  rem[lane + 160] = r5;
  __syncthreads();

  if (lane == 0) {
    int cnt = 0;
    for (int i = 0; i < N_BOXES && cnt < MAX_OUT; ++i) {
      if (((rem[i >> 5] >> (i & 31)) & 1u) == 0u)
        out[cnt++] = order[i];
    }
    for (; cnt < MAX_OUT; ++cnt) out[cnt] = -1;
  }
}

// ---------------------------------------------------------------------------
extern "C" void kernel_launch(void* const* d_in, const int* in_sizes, int n_in,
                              void* d_out, int out_size, void* d_ws, size_t ws_size,
                              hipStream_t stream) {
  (void)in_sizes; (void)n_in; (void)out_size; (void)ws_size;
  const float* rois   = (const float*)d_in[0];   // (N,5) f32
  const float* scores = (const float*)d_in[1];   // (N,1) f32

  char* ws = (char*)d_ws;
  float4*   boxes_s = (float4*)(ws);                               // 98304 B
  float*    area_s  = (float*)(ws + 98304);                        // 24576 B
  int*      order   = (int*)(ws + 98304 + 24576);                  // 24576 B
  uint32_t* mask    = (uint32_t*)(ws + 98304 + 24576 + 24576);     // 4718592 B

  sort_kernel<<<1, 1024, 0, stream>>>(rois, scores, boxes_s, area_s, order);
  mask_kernel<<<dim3(NBLK, NT16), 32, 0, stream>>>(boxes_s, area_s, mask);
  reduce_kernel<<<1, 32, 0, stream>>>(mask, order, (int*)d_out);
}